// KroneckerAddress_11433202942761
// MI455X (gfx1250) — compile-verified
//
#include <hip/hip_runtime.h>
#include <stdint.h>

typedef __attribute__((ext_vector_type(2))) float v2f;
typedef __attribute__((ext_vector_type(8))) float v8f;

#define UFAC     3
#define DPDIM    128
#define TOPK     32
#define NBATCH   64
#define NTHREADS 128

// ---- packed sort keys: positive f32 bits in high word (order-preserving),
// ---- complemented index in low word (ties -> smaller index wins, like jax top_k)
__device__ __forceinline__ uint64_t pack_key(float v, uint32_t idx) {
  return ((uint64_t)__float_as_uint(v) << 32) | (uint64_t)(idx ^ 0xFFFFFFFFu);
}
__device__ __forceinline__ float    key_val(uint64_t k) { return __uint_as_float((uint32_t)(k >> 32)); }
__device__ __forceinline__ uint32_t key_idx(uint64_t k) { return ((uint32_t)k) ^ 0xFFFFFFFFu; }

// ---- in-LDS bitonic sort, descending, n power of two, NTHREADS cooperating threads
__device__ __forceinline__ void bitonic_sort_desc(uint64_t* keys, unsigned n, unsigned t) {
  for (unsigned k = 2; k <= n; k <<= 1) {
    for (unsigned j = k >> 1; j > 0; j >>= 1) {
      for (unsigned p = t; p < (n >> 1); p += NTHREADS) {
        unsigned i       = ((p & ~(j - 1u)) << 1) | (p & (j - 1u));
        unsigned partner = i | j;
        bool     up      = ((i & k) == 0u);     // 'up' segments sort descending
        uint64_t a = keys[i], b = keys[partner];
        if ((a < b) == up) { keys[i] = b; keys[partner] = a; }
      }
      __syncthreads();
    }
  }
}

// ---- 32x32 outer product D = a (32x1) * b (1x32) via 4x V_WMMA_F32_16X16X4_F32
// ---- (one 16x16 tile per wave; K=1 padded to K=4 with zeros).
// ---- Results packed with combined index (aidx*128 + bidx) into outKeys[1024].
__device__ __forceinline__ void outer32_wmma(const uint64_t* __restrict__ ka,
                                             const uint64_t* __restrict__ kb,
                                             uint64_t* __restrict__ outKeys,
                                             unsigned t) {
  const unsigned wave = t >> 5;       // 0..3
  const unsigned lane = t & 31u;
  const unsigned mi   = wave >> 1;    // row tile (0/1)
  const unsigned ni   = wave & 1u;    // col tile (0/1)

  // A 16x4 f32 fragment: lanes 0-15 hold M=lane {K0,K1}; lanes 16-31 {K2,K3}.
  // Only K=0 is non-zero (outer product).
  v2f a; a.x = (lane < 16u) ? key_val(ka[mi * 16u + lane]) : 0.0f; a.y = 0.0f;
  // B 4x16 f32 fragment (row striped across lanes): lanes 0-15 hold N=lane {K0,K1}.
  v2f b; b.x = (lane < 16u) ? key_val(kb[ni * 16u + lane]) : 0.0f; b.y = 0.0f;
  v8f c = {};
  v8f d = __builtin_amdgcn_wmma_f32_16x16x4_f32(
      /*neg_a=*/false, a, /*neg_b=*/false, b,
      /*c_mod=*/(short)0, c, /*reuse_a=*/false, /*reuse_b=*/false);

  // D 16x16 f32 layout: lane<16 -> (M=r, N=lane); lane>=16 -> (M=r+8, N=lane-16).
  const unsigned ncol  = lane & 15u;
  const unsigned mbase = (lane < 16u) ? 0u : 8u;
#pragma unroll
  for (int r = 0; r < 8; ++r) {
    unsigned R = mi * 16u + mbase + (unsigned)r;
    unsigned C = ni * 16u + ncol;
    uint32_t idx = key_idx(ka[R]) * 128u + key_idx(kb[C]);
    outKeys[R * 32u + C] = pack_key(d[r], idx);
  }
}

__global__ __launch_bounds__(NTHREADS)
void kron_topk_kernel(const float* __restrict__ z,
                      const float* __restrict__ log_tau,
                      int*   __restrict__ out_idx,
                      float* __restrict__ out_w) {
  __shared__ float    red[NTHREADS];
  __shared__ uint64_t keys[1024];
  __shared__ uint64_t topF[UFAC][TOPK];
  __shared__ uint64_t topP[TOPK];

  const unsigned t = threadIdx.x;
  const unsigned b = blockIdx.x;
  const float inv_tau = __expf(-log_tau[0]);   // parts / exp(log_tau)

  // ---- per-factor softmax (max-subtracted, like jax.nn.softmax) + top-32 ----
  for (int u = 0; u < UFAC; ++u) {
    float x = z[(size_t)b * (UFAC * DPDIM) + (size_t)u * DPDIM + t] * inv_tau;

    red[t] = x; __syncthreads();
    for (unsigned s = NTHREADS / 2; s > 0; s >>= 1) {
      if (t < s) red[t] = fmaxf(red[t], red[t + s]);
      __syncthreads();
    }
    float m = red[0]; __syncthreads();

    float e = __expf(x - m);
    red[t] = e; __syncthreads();
    for (unsigned s = NTHREADS / 2; s > 0; s >>= 1) {
      if (t < s) red[t] += red[t + s];
      __syncthreads();
    }
    float p = e / red[0]; __syncthreads();

    keys[t] = pack_key(p, t);
    __syncthreads();
    bitonic_sort_desc(keys, DPDIM, t);
    if (t < TOPK) topF[u][t] = keys[t];
    __syncthreads();
  }

  // ---- stage 1: top32(p0) x top32(p1) -> 1024 pair candidates -> top 32 ----
  outer32_wmma(topF[0], topF[1], keys, t);
  __syncthreads();
  bitonic_sort_desc(keys, 1024, t);
  if (t < TOPK) topP[t] = keys[t];
  __syncthreads();

  // ---- stage 2: top32(pairs) x top32(p2) -> 1024 -> final top 32 ----
  // combined index = (i*128 + j)*128 + k = flat index into 128^3 addr space
  outer32_wmma(topP, topF[2], keys, t);
  __syncthreads();
  bitonic_sort_desc(keys, 1024, t);

  if (t < TOPK) {
    uint64_t kk = keys[t];                 // sorted descending, ties -> smaller idx
    out_idx[b * TOPK + t] = (int)key_idx(kk);
    out_w  [b * TOPK + t] = key_val(kk);
  }
}

extern "C" void kernel_launch(void* const* d_in, const int* in_sizes, int n_in,
                              void* d_out, int out_size, void* d_ws, size_t ws_size,
                              hipStream_t stream) {
  const float* z       = (const float*)d_in[0];
  const float* log_tau = (const float*)d_in[1];
  // tuple output: indices (64x32 int32) then weights (64x32 f32), concatenated flat
  int*   out_idx = (int*)d_out;
  float* out_w   = (float*)d_out + NBATCH * TOPK;
  kron_topk_kernel<<<NBATCH, NTHREADS, 0, stream>>>(z, log_tau, out_idx, out_w);
}